// WrappedModel_4947802325590
// MI455X (gfx1250) — compile-verified
//
#include <hip/hip_runtime.h>
#include <math.h>

#define DIMD  128   // input feature dim
#define OUTD  128   // output dim
#define FD    640   // 5*DIMD concatenated features

typedef __attribute__((ext_vector_type(2))) float v2f;
typedef __attribute__((ext_vector_type(8))) float v8f;

// ---------------------------------------------------------------------------
// Kernel 0: repack W[640,128] -> Wt so each lane's B-operand K-pair is a
// contiguous 8B load:  Wt[(k>>1)*256 + col*2 + (k&1)] = W[k*128 + col]
// ---------------------------------------------------------------------------
__global__ void __launch_bounds__(256)
repack_w_kernel(const float* __restrict__ Wm, float* __restrict__ Wt) {
  int idx = blockIdx.x * 256 + threadIdx.x;      // 0 .. 640*128-1
  if (idx >= FD * OUTD) return;
  int k   = idx >> 7;          // / 128
  int col = idx & 127;
  Wt[(size_t)(k >> 1) * (2 * OUTD) + col * 2 + (k & 1)] = Wm[(size_t)k * OUTD + col];
}

// ---------------------------------------------------------------------------
// Kernel 1: per-segment reduction (batch sorted -> contiguous row ranges).
// One block per segment, 128 threads = one per feature column; 512B coalesced
// row streams. This phase is the bandwidth roofline (256 MB @ 23.3 TB/s).
// ---------------------------------------------------------------------------
__global__ void __launch_bounds__(128)
seg_reduce_kernel(const float* __restrict__ x,
                  const int*   __restrict__ batch,
                  float*       __restrict__ feats,
                  int N, int G) {
  const int g = blockIdx.x;
  const int d = threadIdx.x;          // column 0..127

  __shared__ int sBounds[2];
  if (threadIdx.x == 0) {
    int lo = 0, hi = N;
    while (lo < hi) { int mid = (lo + hi) >> 1; if (batch[mid] < g) lo = mid + 1; else hi = mid; }
    sBounds[0] = lo;
    hi = N;
    while (lo < hi) { int mid = (lo + hi) >> 1; if (batch[mid] < g + 1) lo = mid + 1; else hi = mid; }
    sBounds[1] = lo;
  }
  __syncthreads();
  const int start = sBounds[0];
  const int end   = sBounds[1];

  float s  = 0.0f;
  float sq = 0.0f;
  float mn =  INFINITY;   // jax segment_min identity
  float mx = -INFINITY;   // jax segment_max identity

  for (int r = start; r < end; ++r) {
    int pr = r + 8; if (pr >= end) pr = end - 1;
    __builtin_prefetch(&x[(size_t)pr * DIMD + d], 0, 0);   // global_prefetch_b8

    float v = x[(size_t)r * DIMD + d];
    s  += v;
    sq += v * v;
    mn = fminf(mn, v);
    mx = fmaxf(mx, v);
  }

  const float cntRaw = (float)(end - start);
  const float cnt    = cntRaw > 0.0f ? cntRaw : 1.0f;   // jnp.maximum(counts, 1)
  const float mean   = s / cnt;
  const float var    = sq / cnt - mean * mean;
  const float stdv   = sqrtf(fmaxf(var, 1e-9f));
  const float scale  = 0.14142135623730950488f;         // 1/sqrt(AVG_NODES=50)

  const size_t base = (size_t)g * FD;
  feats[base + 0 * DIMD + d] = s * scale;
  feats[base + 1 * DIMD + d] = mean;
  feats[base + 2 * DIMD + d] = mn;
  feats[base + 3 * DIMD + d] = mx;
  feats[base + 4 * DIMD + d] = stdv;
}

// ---------------------------------------------------------------------------
// Kernel 2: out[G,128] = feats[G,640] @ W[640,128] + b via
// V_WMMA_F32_16X16X4_F32. Block = 256 threads (8 waves); block owns one
// 16-row M tile, wave w owns N tile w. K loop: 640/4 = 160 WMMAs, each fed by
// one b64 A load and one b64 (repacked) B load.
//
// f32 A 16x4 per-lane: row = lane&15, lanes 0-15 hold K={0,1}, 16-31 K={2,3}.
// B 4x16 symmetric: col = lane&15, K-pair selected by lane>>4 (repacked
// contiguous). C/D: VGPR i -> M = i + 8*(lane>>4), N = lane&15.
// ---------------------------------------------------------------------------
__global__ void __launch_bounds__(256)
gemm_wmma_kernel(const float* __restrict__ feats,
                 const float* __restrict__ Wt,
                 const float* __restrict__ bias,
                 float*       __restrict__ out,
                 int G) {
  const int wave  = threadIdx.x >> 5;        // 0..7 -> N tile
  const int lane  = threadIdx.x & 31;
  const int mbase = blockIdx.x * 16;
  const int nbase = wave * 16;

  const int rowA  = lane & 15;
  const int colB  = lane & 15;
  const int hiHalf = lane >> 4;              // 0 or 1
  const int kHalf  = hiHalf * 2;             // 0 or 2

  int aRowIdx = mbase + rowA;
  if (aRowIdx > G - 1) aRowIdx = G - 1;      // only matters for partial tile
  const float* aRow  = feats + (size_t)aRowIdx * FD + kHalf;          // 8B aligned
  const float* bBase = Wt + (size_t)hiHalf * (2 * OUTD) + (nbase + colB) * 2;

  v8f acc = {};
  for (int k = 0; k < FD; k += 4) {
    v2f a  = *(const v2f*)(aRow + k);                    // feats K-pair, b64
    v2f bb = *(const v2f*)(bBase + (size_t)k * (OUTD / 2)); // W K-pair, b64
    acc = __builtin_amdgcn_wmma_f32_16x16x4_f32(
        /*neg_a=*/false, a, /*neg_b=*/false, bb,
        /*c_mod=*/(short)0, acc, /*reuse_a=*/false, /*reuse_b=*/false);
  }

  const float bv    = bias[nbase + colB];
  const int   rbase = mbase + hiHalf * 8;
  float* o = out + (size_t)rbase * OUTD + nbase + colB;

  if (mbase + 16 <= G) {
    // Full tile (every tile when G % 16 == 0): straight-line stores, no exec games.
#pragma unroll
    for (int i = 0; i < 8; ++i)
      o[(size_t)i * OUTD] = acc[i] + bv;
  } else {
#pragma unroll
    for (int i = 0; i < 8; ++i)
      if (rbase + i < G)
        o[(size_t)i * OUTD] = acc[i] + bv;
  }
}

// ---------------------------------------------------------------------------
extern "C" void kernel_launch(void* const* d_in, const int* in_sizes, int n_in,
                              void* d_out, int out_size, void* d_ws, size_t ws_size,
                              hipStream_t stream) {
  const float* x     = (const float*)d_in[0];   // [N,128]
  const int*   batch = (const int*)  d_in[1];   // [N] sorted segment ids
  const float* Wm    = (const float*)d_in[2];   // [640,128]
  const float* bias  = (const float*)d_in[3];   // [128]
  // d_in[4] = num_segments (device scalar; G derived from out_size instead)

  const int N = in_sizes[1];
  const int G = out_size / OUTD;

  float* feats = (float*)d_ws;                  // G*640 floats (25.6 MB)
  float* Wt    = feats + (size_t)G * FD;        // 640*128 floats (327 KB)
  float* out   = (float*)d_out;

  repack_w_kernel<<<(FD * OUTD + 255) / 256, 256, 0, stream>>>(Wm, Wt);
  seg_reduce_kernel<<<G, 128, 0, stream>>>(x, batch, feats, N, G);

  const int mTiles = (G + 15) / 16;             // 625 for G=10000
  gemm_wmma_kernel<<<mTiles, 256, 0, stream>>>(feats, Wt, bias, out, G);
}